// RGBDVideoTower_75222057222146
// MI455X (gfx1250) — compile-verified
//
#include <hip/hip_runtime.h>
#include <hip/hip_bf16.h>

// Voxel scatter-mean for MI455X (gfx1250, wave32).
// Bandwidth-bound (AI ~0.07 FLOP/B): no matrix ops exist in this workload, so
// the optimal path is b128 vmem + native L2 f32 atomics (confirmed lowering in
// round 1: vmem=21, low valu => no CAS loops). The normalize pass (largest
// traffic, ~1 GB) uses the CDNA5 async global->LDS datapath (ASYNCcnt).

#define GRID_DIM 40
#define NUM_VOX (GRID_DIM * GRID_DIM * GRID_DIM)   // 64000

typedef float v4f __attribute__((ext_vector_type(4)));
typedef int vox_b128 __attribute__((vector_size(16)));   // matches builtin param

#define AS1 __attribute__((address_space(1)))
#define AS3 __attribute__((address_space(3)))

#if defined(__gfx1250__) && __has_builtin(__builtin_amdgcn_global_load_async_to_lds_b128)
#define VOX_ASYNC_LDS 1
#else
#define VOX_ASYNC_LDS 0
#endif

// ---------------------------------------------------------------------------
// K0: zero pooled output (b128 stores) and counts/max workspace.
// ---------------------------------------------------------------------------
__global__ void voxpool_init_kernel(float* __restrict__ out,
                                    unsigned* __restrict__ cnt,
                                    long n_quads, int n_ints) {
    long tid    = (long)blockIdx.x * blockDim.x + threadIdx.x;
    long stride = (long)gridDim.x * blockDim.x;
    v4f* o4 = (v4f*)out;
    v4f z = (v4f){0.0f, 0.0f, 0.0f, 0.0f};
    for (long i = tid; i < n_quads; i += stride) o4[i] = z;
    for (long i = tid; i < (long)n_ints; i += stride) cnt[i] = 0u;
}

// ---------------------------------------------------------------------------
// K1: per point -> validity, voxel id, count atomic, per-batch max atomic.
// Matches JAX bit-for-bit: strict bounds, masked xyz, (x - min)/0.025f, clip.
// ---------------------------------------------------------------------------
__global__ void voxpool_ids_kernel(const float* __restrict__ xyz,
                                   int* __restrict__ codes,
                                   unsigned* __restrict__ cnt,
                                   int* __restrict__ maxv,
                                   int N, int total) {
    int p = blockIdx.x * blockDim.x + threadIdx.x;
    if (p >= total) return;
    int b = p / N;

    float x = xyz[3 * p + 0];
    float y = xyz[3 * p + 1];
    float z = xyz[3 * p + 2];

    bool valid = (x > -0.5f) && (x < 0.5f) &&
                 (y > -0.5f) && (y < 0.5f) &&
                 (z >  0.0f) && (z < 1.0f);

    float xm = valid ? x : 0.0f;
    float ym = valid ? y : 0.0f;
    float zm = valid ? z : 0.0f;

    int ix = (int)floorf((xm + 0.5f) / 0.025f);
    int iy = (int)floorf((ym + 0.5f) / 0.025f);
    int iz = (int)floorf((zm - 0.0f) / 0.025f);
    ix = min(max(ix, 0), GRID_DIM - 1);
    iy = min(max(iy, 0), GRID_DIM - 1);
    iz = min(max(iz, 0), GRID_DIM - 1);
    int vid = ix + GRID_DIM * iy + GRID_DIM * GRID_DIM * iz;

    // Every point (valid or not) increments its segment count, per reference.
    atomicAdd(&cnt[(long)b * NUM_VOX + vid], 1u);
    atomicMax(&maxv[b], vid);

    // Sign bit marks invalid (zero feature contribution -> skip scatter).
    codes[p] = valid ? vid : (vid | 0x80000000);
}

// ---------------------------------------------------------------------------
// K2: scatter features. One 256-thread block (8 waves) per point.
// Valid points: 256 lanes x (1 non-temporal b128 load + 4 f32 L2 atomics).
// Invalid points contribute zeros -> whole block exits (also removes the
// contention hot-spot at the masked voxel).
// ---------------------------------------------------------------------------
__global__ void voxpool_scatter_kernel(const float* __restrict__ feats,
                                       const int* __restrict__ codes,
                                       float* __restrict__ out,
                                       int N, int C) {
    int p = blockIdx.x;              // point index in [0, B*N)
    int code = codes[p];             // block-uniform -> scalar load
    if (code < 0) return;            // invalid point: zero contribution
    int b = p / N;

    const v4f* f4 = (const v4f*)(feats + (long)p * C);
    float* dst = out + ((long)b * NUM_VOX + code) * C;

    int nq = C >> 2;                 // 256 quads for C=1024
    for (int q = threadIdx.x; q < nq; q += blockDim.x) {
        v4f v = __builtin_nontemporal_load(&f4[q]);   // stream-once read
        float* d = dst + 4 * q;
        __hip_atomic_fetch_add(&d[0], v.x, __ATOMIC_RELAXED, __HIP_MEMORY_SCOPE_AGENT);
        __hip_atomic_fetch_add(&d[1], v.y, __ATOMIC_RELAXED, __HIP_MEMORY_SCOPE_AGENT);
        __hip_atomic_fetch_add(&d[2], v.z, __ATOMIC_RELAXED, __HIP_MEMORY_SCOPE_AGENT);
        __hip_atomic_fetch_add(&d[3], v.w, __ATOMIC_RELAXED, __HIP_MEMORY_SCOPE_AGENT);
    }
}

// ---------------------------------------------------------------------------
// K3: normalize: out[v, :] *= 1 / max(count[v], 1). One block per voxel.
// gfx1250 path: async global->LDS b128 (ASYNCcnt), wait, ds_load readback of
// the lane-private slot, scale, non-temporal b128 store (output is written
// once and never re-read on device -> keep it out of L2).
// ---------------------------------------------------------------------------
__global__ void voxpool_norm_kernel(float* __restrict__ out,
                                    const unsigned* __restrict__ cnt,
                                    int C) {
    __shared__ __align__(16) float tile[1024];
    long v = blockIdx.x;             // b*NUM_VOX + vox
    float c = fmaxf((float)cnt[v], 1.0f);   // uniform -> scalar load
    float inv = 1.0f / c;
    float* row = out + v * (long)C;
    int nq = C >> 2;
    for (int q = threadIdx.x; q < nq; q += blockDim.x) {
        float* g = row + 4 * q;
#if VOX_ASYNC_LDS
        float* l = tile + 4 * q;     // lane-private LDS slot
        __builtin_amdgcn_global_load_async_to_lds_b128(
            (AS1 vox_b128*)g, (AS3 vox_b128*)l, /*offset=*/0, /*cpol=*/0);
        asm volatile("s_wait_asynccnt 0x0" ::: "memory");
        v4f val = *(const v4f*)l;    // ds_load_b128 of own slot (no barrier)
#else
        v4f val = *(const v4f*)g;
#endif
        __builtin_nontemporal_store(val * inv, (v4f*)g);
    }
}

// ---------------------------------------------------------------------------
// K4: batch_offset = cumsum(max_id_per_batch + 1), appended after pooled.
// ---------------------------------------------------------------------------
__global__ void voxpool_offset_kernel(const int* __restrict__ maxv,
                                      float* __restrict__ out_off, int B) {
    if (blockIdx.x == 0 && threadIdx.x == 0) {
        int acc = 0;
        for (int b = 0; b < B; ++b) {
            acc += maxv[b] + 1;
            out_off[b] = (float)acc;
        }
    }
}

extern "C" void kernel_launch(void* const* d_in, const int* in_sizes, int n_in,
                              void* d_out, int out_size, void* d_ws, size_t ws_size,
                              hipStream_t stream) {
    (void)n_in; (void)ws_size;
    const float* feats = (const float*)d_in[0];   // (B, N, C) f32
    const float* xyz   = (const float*)d_in[1];   // (B, N, 3) f32
    float* out = (float*)d_out;                   // pooled (B,NUM_VOX,C) ++ batch_offset (B)

    // Derive shapes: out_size = B*(NUM_VOX*C) + B.
    int BN = in_sizes[1] / 3;                     // B*N = 36864
    int C  = in_sizes[0] / BN;                    // 1024
    int B  = out_size / (NUM_VOX * C + 1);        // 2
    int N  = BN / B;                              // 18432

    // Workspace layout: [counts: B*NUM_VOX u32][maxv: B i32][codes: B*N i32]
    unsigned* cnt = (unsigned*)d_ws;
    int* maxv  = (int*)(cnt + (long)B * NUM_VOX);
    int* codes = maxv + B;

    long pooled  = (long)B * NUM_VOX * C;         // 131,072,000
    long n_quads = pooled >> 2;
    int  n_ints  = B * NUM_VOX + B;               // counts + maxv (contiguous)

    voxpool_init_kernel<<<8192, 256, 0, stream>>>(out, cnt, n_quads, n_ints);
    voxpool_ids_kernel<<<(BN + 255) / 256, 256, 0, stream>>>(xyz, codes, cnt, maxv, N, BN);
    voxpool_scatter_kernel<<<BN, 256, 0, stream>>>(feats, codes, out, N, C);
    voxpool_norm_kernel<<<B * NUM_VOX, 256, 0, stream>>>(out, cnt, C);
    voxpool_offset_kernel<<<1, 32, 0, stream>>>(maxv, out + pooled, B);
}